// LightGCNEncoder_43061342110015
// MI455X (gfx1250) — compile-verified
//
#include <hip/hip_runtime.h>
#include <hip/hip_bf16.h>

#define N_PROJECT 100000
#define N_COMPANY 50000
#define N_EDGES   1000000
#define INPUT_DIM 256
#define HIDDEN_DIM 128

typedef float v2f __attribute__((ext_vector_type(2)));
typedef float v8f __attribute__((ext_vector_type(8)));

// ---------------------------------------------------------------------------
// Zero a float4-aligned region
// ---------------------------------------------------------------------------
__global__ void zero_f4_kernel(float4* __restrict__ p, size_t n4) {
    size_t i = (size_t)blockIdx.x * blockDim.x + threadIdx.x;
    if (i < n4) p[i] = make_float4(0.f, 0.f, 0.f, 0.f);
}

// ---------------------------------------------------------------------------
// Degree counting for both relations into one segmented u32 array:
//   [0,Np): pc_src deg  [Np,Np+Nc): pc_dst deg
//   [Np+Nc, Np+2Nc): cp_src deg  [Np+2Nc, 2Np+2Nc): cp_dst deg
// ---------------------------------------------------------------------------
__global__ void degree_kernel(const int* __restrict__ pc_src, const int* __restrict__ pc_dst,
                              const int* __restrict__ cp_src, const int* __restrict__ cp_dst,
                              unsigned int* __restrict__ deg, int nEdges) {
    int e = blockIdx.x * blockDim.x + threadIdx.x;
    if (e >= nEdges) return;
    atomicAdd(&deg[pc_src[e]], 1u);
    atomicAdd(&deg[N_PROJECT + pc_dst[e]], 1u);
    atomicAdd(&deg[N_PROJECT + N_COMPANY + cp_src[e]], 1u);
    atomicAdd(&deg[N_PROJECT + 2 * N_COMPANY + cp_dst[e]], 1u);
}

__global__ void invsqrt_kernel(const unsigned int* __restrict__ deg,
                               float* __restrict__ inv, int n) {
    int i = blockIdx.x * blockDim.x + threadIdx.x;
    if (i >= n) return;
    unsigned int d = deg[i];
    inv[i] = (d > 0u) ? rsqrtf((float)d) : 0.0f;
}

// ---------------------------------------------------------------------------
// Fused GEMM + bias using fp32 WMMA:  H = X @ W + b   (also copies into ACC)
//   X: [M, 256] row-major, W: [256, 128] row-major, b: [128]
// One 256-thread block (8 waves) computes a 16-row x 128-col strip:
// wave w owns the 16x16 tile at columns [16w, 16w+16).
// A frag (16x4 f32): lanes 0-15 hold M=lane K={k0,k0+1}; lanes 16-31 K={k0+2,k0+3}
// B frag (4x16 f32): lanes 0-15 hold N=lane K={k0,k0+1}; lanes 16-31 K={k0+2,k0+3}
// C/D (16x16 f32):   vgpr r, lanes 0-15 -> (M=r,     N=lane)
//                            lanes16-31 -> (M=r+8,   N=lane-16)
// M is an exact multiple of 16, so EXEC is all-1s for every WMMA.
// TWO independent accumulators (even/odd K-steps) break the 64-deep WMMA RAW
// chain so the XDL pipe can overlap two v_wmma streams; merged at the end.
// ---------------------------------------------------------------------------
__global__ void gemm_bias_wmma_kernel(const float* __restrict__ X,
                                      const float* __restrict__ W,
                                      const float* __restrict__ bias,
                                      float* __restrict__ H,
                                      float* __restrict__ ACC) {
    const int wave = threadIdx.x >> 5;     // 0..7
    const int lane = threadIdx.x & 31;
    const int half = lane >> 4;            // 0 or 1
    const int l16  = lane & 15;
    const int row0 = blockIdx.x * 16;
    const int bcol = wave * 16 + l16;      // this lane's N (B frag and C/D frag)

    const float* __restrict__ Arow = X + (size_t)(row0 + l16) * INPUT_DIM;

    v8f c0 = {};
    v8f c1 = {};
    #pragma unroll 4
    for (int k0 = 0; k0 < INPUT_DIM; k0 += 8) {
        // even K-step -> c0
        {
            const int ka = k0 + 2 * half;
            v2f a, b;
            a.x = Arow[ka];
            a.y = Arow[ka + 1];
            b.x = W[(size_t)ka * HIDDEN_DIM + bcol];
            b.y = W[(size_t)(ka + 1) * HIDDEN_DIM + bcol];
            c0 = __builtin_amdgcn_wmma_f32_16x16x4_f32(false, a, false, b,
                                                       (short)0, c0, false, false);
        }
        // odd K-step -> c1 (independent of c0 chain)
        {
            const int ka = k0 + 4 + 2 * half;
            v2f a, b;
            a.x = Arow[ka];
            a.y = Arow[ka + 1];
            b.x = W[(size_t)ka * HIDDEN_DIM + bcol];
            b.y = W[(size_t)(ka + 1) * HIDDEN_DIM + bcol];
            c1 = __builtin_amdgcn_wmma_f32_16x16x4_f32(false, a, false, b,
                                                       (short)0, c1, false, false);
        }
    }

    const float bv = bias[bcol];
    #pragma unroll
    for (int r = 0; r < 8; ++r) {
        const int m = row0 + r + 8 * half;
        const float v = c0[r] + c1[r] + bv;
        const size_t idx = (size_t)m * HIDDEN_DIM + bcol;
        H[idx]   = v;
        ACC[idx] = v;
    }
}

// ---------------------------------------------------------------------------
// Edge-parallel normalized scatter-add: one wave32 per edge, float4 per lane.
//   Out[dst] += Hsrc[src] * (inv_s[src] * inv_d[dst])
// Gathers + f32 atomics both resolve in the 192MB L2 (h arrays are 26-51MB).
// ---------------------------------------------------------------------------
__global__ void scatter_edges_kernel(const float* __restrict__ Hsrc,
                                     const int* __restrict__ src,
                                     const int* __restrict__ dst,
                                     const float* __restrict__ inv_s,
                                     const float* __restrict__ inv_d,
                                     float* __restrict__ Out, int nEdges) {
    const int gid  = blockIdx.x * blockDim.x + threadIdx.x;
    const int e    = gid >> 5;
    const int lane = gid & 31;
    if (e >= nEdges) return;
    const int s = src[e];
    const int d = dst[e];
    const float w = inv_s[s] * inv_d[d];
    const float4 v = ((const float4*)(Hsrc + (size_t)s * HIDDEN_DIM))[lane];
    float* op = Out + (size_t)d * HIDDEN_DIM + lane * 4;
    unsafeAtomicAdd(op + 0, v.x * w);   // global_atomic_add_f32
    unsafeAtomicAdd(op + 1, v.y * w);
    unsafeAtomicAdd(op + 2, v.z * w);
    unsafeAtomicAdd(op + 3, v.w * w);
}

// ---------------------------------------------------------------------------
// h = new;  acc += new   (float4 vectorized)
// ---------------------------------------------------------------------------
__global__ void layer_update_kernel(float4* __restrict__ H, float4* __restrict__ ACC,
                                    const float4* __restrict__ NEW, size_t n4) {
    size_t i = (size_t)blockIdx.x * blockDim.x + threadIdx.x;
    if (i >= n4) return;
    const float4 v = NEW[i];
    H[i] = v;
    float4 a = ACC[i];
    a.x += v.x; a.y += v.y; a.z += v.z; a.w += v.w;
    ACC[i] = a;
}

// ---------------------------------------------------------------------------
// out = (acc/3) / max(||acc/3||_2, 1e-12)  per 128-wide row; one wave per row,
// wave32 shuffle reduction (4 f32 per lane).
// ---------------------------------------------------------------------------
__global__ void normalize_rows_kernel(float* __restrict__ Out, int nRows) {
    const int wave = threadIdx.x >> 5;
    const int lane = threadIdx.x & 31;
    const int row  = blockIdx.x * 8 + wave;
    if (row >= nRows) return;
    float4* p = (float4*)(Out + (size_t)row * HIDDEN_DIM);
    float4 v = p[lane];
    const float s = 1.0f / 3.0f;           // 1/(NUM_LAYERS+1)
    v.x *= s; v.y *= s; v.z *= s; v.w *= s;
    float ss = v.x * v.x + v.y * v.y + v.z * v.z + v.w * v.w;
    #pragma unroll
    for (int off = 16; off > 0; off >>= 1) ss += __shfl_xor(ss, off, 32);
    const float f = 1.0f / fmaxf(sqrtf(ss), 1e-12f);
    v.x *= f; v.y *= f; v.z *= f; v.w *= f;
    p[lane] = v;
}

// ---------------------------------------------------------------------------
// Host orchestration
// ---------------------------------------------------------------------------
extern "C" void kernel_launch(void* const* d_in, const int* in_sizes, int n_in,
                              void* d_out, int out_size, void* d_ws, size_t ws_size,
                              hipStream_t stream) {
    (void)in_sizes; (void)n_in; (void)out_size; (void)ws_size;

    const float* x_project = (const float*)d_in[0];
    const float* x_company = (const float*)d_in[1];
    const float* W_project = (const float*)d_in[2];
    const float* b_project = (const float*)d_in[3];
    const float* W_company = (const float*)d_in[4];
    const float* b_company = (const float*)d_in[5];
    const int*   pc_src    = (const int*)d_in[6];
    const int*   pc_dst    = (const int*)d_in[7];
    const int*   cp_src    = (const int*)d_in[8];
    const int*   cp_dst    = (const int*)d_in[9];

    const size_t nP = (size_t)N_PROJECT * HIDDEN_DIM;   // 12.8M floats
    const size_t nC = (size_t)N_COMPANY * HIDDEN_DIM;   //  6.4M floats
    const int    nDeg = 2 * (N_PROJECT + N_COMPANY);    // 300000

    // Workspace layout (floats)
    float* ws     = (float*)d_ws;
    float* h_p    = ws;                     // [Np,128]
    float* h_c    = h_p + nP;               // [Nc,128]
    float* new_p  = h_c + nC;               // [Np,128]
    float* new_c  = new_p + nP;             // [Nc,128]
    float* inv    = new_c + nC;             // [300000]
    unsigned int* deg = (unsigned int*)(inv + nDeg);    // [300000]

    const float* inv_pc_s = inv;                                 // projects
    const float* inv_pc_d = inv + N_PROJECT;                     // companies
    const float* inv_cp_s = inv + N_PROJECT + N_COMPANY;         // companies
    const float* inv_cp_d = inv + N_PROJECT + 2 * N_COMPANY;     // projects

    // Accumulators live directly in d_out: [acc_p | acc_c]
    float* acc_p = (float*)d_out;
    float* acc_c = acc_p + nP;

    const int TB = 256;

    // 1) degrees -> inverse sqrt
    zero_f4_kernel<<<(nDeg / 4 + TB - 1) / TB, TB, 0, stream>>>((float4*)deg, nDeg / 4);
    degree_kernel<<<(N_EDGES + TB - 1) / TB, TB, 0, stream>>>(pc_src, pc_dst, cp_src, cp_dst,
                                                              deg, N_EDGES);
    invsqrt_kernel<<<(nDeg + TB - 1) / TB, TB, 0, stream>>>(deg, inv, nDeg);

    // 2) input projections (WMMA fp32), acc initialized to h
    gemm_bias_wmma_kernel<<<N_PROJECT / 16, 256, 0, stream>>>(x_project, W_project, b_project,
                                                              h_p, acc_p);
    gemm_bias_wmma_kernel<<<N_COMPANY / 16, 256, 0, stream>>>(x_company, W_company, b_company,
                                                              h_c, acc_c);

    // 3) two propagation layers
    const size_t nAll4 = (nP + nC) / 4;      // new_p|new_c are contiguous
    const int edgeBlocks = (int)(((size_t)N_EDGES * 32 + TB - 1) / TB);
    for (int layer = 0; layer < 2; ++layer) {
        zero_f4_kernel<<<(int)((nAll4 + TB - 1) / TB), TB, 0, stream>>>((float4*)new_p, nAll4);
        // new_p <- aggregate h_c over cp edges (company -> project)
        scatter_edges_kernel<<<edgeBlocks, TB, 0, stream>>>(h_c, cp_src, cp_dst,
                                                            inv_cp_s, inv_cp_d, new_p, N_EDGES);
        // new_c <- aggregate h_p over pc edges (project -> company)
        scatter_edges_kernel<<<edgeBlocks, TB, 0, stream>>>(h_p, pc_src, pc_dst,
                                                            inv_pc_s, inv_pc_d, new_c, N_EDGES);
        // h = new; acc += new
        layer_update_kernel<<<(int)((nP / 4 + TB - 1) / TB), TB, 0, stream>>>(
            (float4*)h_p, (float4*)acc_p, (const float4*)new_p, nP / 4);
        layer_update_kernel<<<(int)((nC / 4 + TB - 1) / TB), TB, 0, stream>>>(
            (float4*)h_c, (float4*)acc_c, (const float4*)new_c, nC / 4);
    }

    // 4) scale by 1/3 and L2-normalize each 128-wide row (both outputs contiguous)
    const int nRows = N_PROJECT + N_COMPANY;   // 150000, divisible by 8
    normalize_rows_kernel<<<nRows / 8, 256, 0, stream>>>(acc_p, nRows);
}